// FraudDetectionHybrid_65481071410614
// MI455X (gfx1250) — compile-verified
//
#include <hip/hip_runtime.h>
#include <stdint.h>

#ifndef __has_builtin
#define __has_builtin(x) 0
#endif

#define LNUM 9
#define BLOCK 256
#define RPT 16                           // rows per thread
#define ROWS_PER_BLOCK (BLOCK * RPT)     // 4096 rows -> 32 KB of x per block

typedef float v2f __attribute__((ext_vector_type(2)));
typedef int   v4i __attribute__((vector_size(16)));

// ---------------------------------------------------------------------------
// Folded constants (written to d_ws by fold_consts, read by the main kernel)
// ---------------------------------------------------------------------------
struct Consts {
    float W[LNUM][4];   // [W00,W01,W10,W11] per block
    float B[LNUM][2];   // bias
    float A[LNUM][2];   // post-relu scale  = rsqrt(var+eps)*gamma*scale
    float C[LNUM][2];   // post-relu offset = (beta - mean*rs*gamma)*scale + shift
    float W1[4][2];
    float B1[4];
    float DW2[4];       // W2[0,:] - W2[1,:]
    float DB2;          // b2[0] - b2[1]
    float DWF;          // Wf[0] - Wf[1]
    float CF;           // Wf[1] + bf
};

// ---------------------------------------------------------------------------
// Fast math helpers (native gfx1250 transcendentals)
// ---------------------------------------------------------------------------
__device__ __forceinline__ float frcp_(float x) { return __builtin_amdgcn_rcpf(x); }

__device__ __forceinline__ float ftanh_(float x) {
#if __has_builtin(__builtin_amdgcn_tanhf)
    return __builtin_amdgcn_tanhf(x);          // v_tanh_f32 (gfx1250)
#else
    float e = __expf(2.0f * x);                // v_exp_f32
    return fmaf(-2.0f, frcp_(e + 1.0f), 1.0f);
#endif
}

__device__ __forceinline__ float fsigmoid_(float x) {
    return frcp_(1.0f + __expf(-x));           // v_exp_f32 + v_rcp_f32
}

// ---------------------------------------------------------------------------
// Two independent rows processed in lockstep as <2 x float> lanes:
// lane .x = row A, lane .y = row B.  All constants are scalar broadcasts, so
// fused multiply-adds are eligible for v_pk_fma_f32 / VOPD dual-issue.
//   h0 = {xA0, xB0}, h1 = {xA1, xB1}
// ---------------------------------------------------------------------------
__device__ __forceinline__ v2f rowpair_forward(v2f h0, v2f h1, const Consts& K) {
#pragma unroll
    for (int l = 0; l < LNUM; ++l) {
        v2f z0 = h0 * K.W[l][0] + (h1 * K.W[l][1] + K.B[l][0]);
        v2f z1 = h0 * K.W[l][2] + (h1 * K.W[l][3] + K.B[l][1]);
        z0 = __builtin_elementwise_max(z0, (v2f){0.0f, 0.0f});
        z1 = __builtin_elementwise_max(z1, (v2f){0.0f, 0.0f});
        h0 = z0 * K.A[l][0] + K.C[l][0];
        h1 = z1 * K.A[l][1] + K.C[l][1];
    }
    // l0 - l1 = sum_i tanh(u_i) * DW2_i + DB2  ;  p0 = sigmoid(l0 - l1)
    v2f acc = {K.DB2, K.DB2};
#pragma unroll
    for (int i = 0; i < 4; ++i) {
        v2f u = h0 * K.W1[i][0] + (h1 * K.W1[i][1] + K.B1[i]);
        v2f t = {ftanh_(u.x), ftanh_(u.y)};                        // v_tanh_f32
        acc = t * K.DW2[i] + acc;
    }
    v2f p = {fsigmoid_(acc.x), fsigmoid_(acc.y)};
    v2f o = p * K.DWF + K.CF;
    return (v2f){fsigmoid_(o.x), fsigmoid_(o.y)};
}

__device__ __forceinline__ float row_forward(float h0, float h1, const Consts& K) {
    v2f r = rowpair_forward((v2f){h0, h0}, (v2f){h1, h1}, K);
    return r.x;
}

// ---------------------------------------------------------------------------
// Async global -> LDS staging (gfx1250 ASYNCcnt path)
// ---------------------------------------------------------------------------
__device__ __forceinline__ void async_cp16(const float4* g, float4* l) {
#if __has_builtin(__builtin_amdgcn_global_load_async_to_lds_b128)
    __builtin_amdgcn_global_load_async_to_lds_b128((v4i*)g, (v4i*)l, 0, 0);
#else
    unsigned lds_off = (unsigned)(uintptr_t)l;   // low 32 bits == LDS byte offset
    asm volatile("global_load_async_to_lds_b128 %0, %1, off"
                 :: "v"(lds_off), "v"(g) : "memory");
#endif
}

__device__ __forceinline__ void wait_async0() {
#if __has_builtin(__builtin_amdgcn_s_wait_asynccnt)
    __builtin_amdgcn_s_wait_asynccnt(0);
#else
    asm volatile("s_wait_asynccnt 0" ::: "memory");
#endif
}

// ---------------------------------------------------------------------------
// Main streaming kernel: 4096 rows / block, x async-staged through 32 KB LDS.
// No forced unroll: each iteration already carries 2 independent packed row
// chains; keeping register pressure moderate preserves resident waves, which
// is what hides ds/VALU latency in a streaming kernel.
// ---------------------------------------------------------------------------
__global__ __launch_bounds__(BLOCK) void fraud_fwd(const float* __restrict__ x,
                                                   float* __restrict__ out,
                                                   const Consts* __restrict__ cp,
                                                   long nrows) {
    __shared__ float4 xs[ROWS_PER_BLOCK / 2];   // 2048 float4 = 32 KB
    const int  tid      = threadIdx.x;
    const long base_row = (long)blockIdx.x * ROWS_PER_BLOCK;

    if (base_row + ROWS_PER_BLOCK <= nrows) {
        // ---- async-stage 32 KB of x into LDS (coalesced b128, ASYNCcnt) ----
        const float4* gx = (const float4*)x + (base_row >> 1);
#pragma unroll
        for (int i = 0; i < RPT / 2; ++i) {
            int idx = tid + i * BLOCK;          // lane-contiguous: 512 B / wave-issue
            async_cp16(gx + idx, &xs[idx]);
        }
        wait_async0();
        __syncthreads();

        const Consts K = *cp;                   // uniform -> s_load, SGPR-resident
        for (int s = 0; s < RPT / 2; ++s) {
            int    fidx = tid + s * BLOCK;      // one float4 == 2 rows
            float4 v    = xs[fidx];             // ds_load_b128
            v2f h0 = {v.x, v.z};                // {rowA.x0, rowB.x0}
            v2f h1 = {v.y, v.w};                // {rowA.x1, rowB.x1}
            v2f r  = rowpair_forward(h0, h1, K);
            *(v2f*)(out + base_row + (long)fidx * 2) = r;   // global_store_b64
        }
    } else {
        // ---- tail block (unused for B = 2^23, kept for generality) ----
        const Consts K = *cp;
        for (int i = tid; i < ROWS_PER_BLOCK; i += BLOCK) {
            long r = base_row + i;
            if (r < nrows) {
                out[r] = row_forward(x[r * 2 + 0], x[r * 2 + 1], K);
            }
        }
    }
}

// ---------------------------------------------------------------------------
// One-wave setup kernel: fold BN/affine/softmax/final-linear constants
// ---------------------------------------------------------------------------
__global__ void fold_consts(const float* __restrict__ W,  const float* __restrict__ b,
                            const float* __restrict__ g,  const float* __restrict__ be,
                            const float* __restrict__ m,  const float* __restrict__ v,
                            const float* __restrict__ sc, const float* __restrict__ sh,
                            const float* __restrict__ W1, const float* __restrict__ b1,
                            const float* __restrict__ W2, const float* __restrict__ b2,
                            const float* __restrict__ Wf, const float* __restrict__ bf,
                            Consts* __restrict__ out) {
    if (threadIdx.x != 0 || blockIdx.x != 0) return;
    for (int l = 0; l < LNUM; ++l) {
        for (int j = 0; j < 2; ++j) {
            out->W[l][j * 2 + 0] = W[l * 4 + j * 2 + 0];
            out->W[l][j * 2 + 1] = W[l * 4 + j * 2 + 1];
            out->B[l][j]         = b[l * 2 + j];
            float s = __frsqrt_rn(v[l * 2 + j] + 1e-5f) * g[l * 2 + j];
            out->A[l][j] = s * sc[l * 2 + j];
            out->C[l][j] = (be[l * 2 + j] - m[l * 2 + j] * s) * sc[l * 2 + j] + sh[l * 2 + j];
        }
    }
    for (int i = 0; i < 4; ++i) {
        out->W1[i][0] = W1[i * 2 + 0];
        out->W1[i][1] = W1[i * 2 + 1];
        out->B1[i]    = b1[i];
        out->DW2[i]   = W2[i] - W2[4 + i];   // W2 is [2,4] row-major
    }
    out->DB2 = b2[0] - b2[1];
    out->DWF = Wf[0] - Wf[1];
    out->CF  = Wf[1] + bf[0];
}

// ---------------------------------------------------------------------------
// Launch
// ---------------------------------------------------------------------------
extern "C" void kernel_launch(void* const* d_in, const int* in_sizes, int n_in,
                              void* d_out, int out_size, void* d_ws, size_t ws_size,
                              hipStream_t stream) {
    const float* x  = (const float*)d_in[0];
    const float* W  = (const float*)d_in[1];
    const float* b  = (const float*)d_in[2];
    const float* g  = (const float*)d_in[3];
    const float* be = (const float*)d_in[4];
    const float* m  = (const float*)d_in[5];
    const float* v  = (const float*)d_in[6];
    const float* sc = (const float*)d_in[7];
    const float* sh = (const float*)d_in[8];
    const float* W1 = (const float*)d_in[9];
    const float* b1 = (const float*)d_in[10];
    const float* W2 = (const float*)d_in[11];
    const float* b2 = (const float*)d_in[12];
    const float* Wf = (const float*)d_in[13];
    const float* bf = (const float*)d_in[14];

    Consts* cp = (Consts*)d_ws;
    fold_consts<<<1, 32, 0, stream>>>(W, b, g, be, m, v, sc, sh,
                                      W1, b1, W2, b2, Wf, bf, cp);

    long nrows   = (long)in_sizes[0] / 2;
    int  nblocks = (int)((nrows + ROWS_PER_BLOCK - 1) / ROWS_PER_BLOCK);
    fraud_fwd<<<nblocks, BLOCK, 0, stream>>>(x, (float*)d_out, cp, nrows);
}